// GNNModel_88399016886916
// MI455X (gfx1250) — compile-verified
//
#include <hip/hip_runtime.h>
#include <math.h>

typedef float v2f __attribute__((ext_vector_type(2)));
typedef float v8f __attribute__((ext_vector_type(8)));

#define HID 64

// ---------------------------------------------------------------- init: deg=1 (self loop), s=0, g_accum=0
__global__ void k_init(float* deg, float* s, float* gacc, int N) {
    int gid = blockIdx.x * blockDim.x + threadIdx.x;
    if (gid < N) { deg[gid] = 1.0f; s[gid] = 0.0f; }
    if (gid < HID) gacc[gid] = 0.0f;
}

// ---------------------------------------------------------------- degree over real edges (col side)
__global__ void k_degree(const int* __restrict__ col, float* __restrict__ deg, int E) {
    int gid = blockIdx.x * blockDim.x + threadIdx.x;
    if (gid < E) atomicAdd(&deg[col[gid]], 1.0f);
}

// ---------------------------------------------------------------- deg -> deg^{-1/2} in place (deg >= 1 always)
__global__ void k_dinv(float* d, int N) {
    int gid = blockIdx.x * blockDim.x + threadIdx.x;
    if (gid < N) d[gid] = rsqrtf(d[gid]);
}

// ---------------------------------------------------------------- layer-1 scalar edge scatter: s[c] += dinv[r]*dinv[c]*x[r]
__global__ void k_l1_scatter(const int* __restrict__ row, const int* __restrict__ col,
                             const float* __restrict__ x, const float* __restrict__ dinv,
                             float* __restrict__ s, int E) {
    int gid = blockIdx.x * blockDim.x + threadIdx.x;
    if (gid >= E) return;
    int r = row[gid], c = col[gid];
    atomicAdd(&s[c], dinv[r] * dinv[c] * x[r]);
}

// ---------------------------------------------------------------- h1[n][j] = relu((s[n] + dinv[n]^2 x[n]) * W1[j] + b1[j])
__global__ void k_l1_finish(const float* __restrict__ s, const float* __restrict__ dinv,
                            const float* __restrict__ x, const float* __restrict__ W1,
                            const float* __restrict__ b1, float* __restrict__ h1, int N) {
    int gid = blockIdx.x * blockDim.x + threadIdx.x;
    if (gid >= N * HID) return;
    int n = gid >> 6, j = gid & 63;
    float dn = dinv[n];
    float sv = s[n] + dn * dn * x[n];
    float h = sv * W1[j] + b1[j];
    h1[gid] = fmaxf(h, 0.0f);
}

// ---------------------------------------------------------------- t = h1 @ W2 via V_WMMA_F32_16X16X4_F32
// one wave per 16-row tile; A slab held in regs; loops 4 col-tiles of 16.
__global__ void k_gemm_wmma(const float* __restrict__ h1, const float* __restrict__ W2,
                            float* __restrict__ t, int N) {
    int lane = threadIdx.x & 31;
    int wid  = (blockIdx.x * blockDim.x + threadIdx.x) >> 5;
    int ntiles = (N + 15) >> 4;
    if (wid >= ntiles) return;
    int base = wid << 4;
    int half = lane >> 4;     // 0: lanes 0-15, 1: lanes 16-31
    int ml   = lane & 15;

    if (base + 16 <= N) {
        // A 16x4 layout per step s (K = 4s..4s+3): a.x = A[m][2*half+4s], a.y = +1
        v2f a[16];
        const float* ap = h1 + (size_t)(base + ml) * HID + 2 * half;
        #pragma unroll
        for (int s = 0; s < 16; ++s) { a[s].x = ap[4 * s]; a[s].y = ap[4 * s + 1]; }

        #pragma unroll
        for (int ct = 0; ct < 4; ++ct) {
            int ncol = ct * 16;
            v8f c = {0.f, 0.f, 0.f, 0.f, 0.f, 0.f, 0.f, 0.f};
            const float* bp = W2 + (size_t)(2 * half) * HID + ncol + ml;
            #pragma unroll
            for (int s = 0; s < 16; ++s) {
                v2f b;
                b.x = bp[(4 * s) * HID];       // B[k=4s+2*half][n]
                b.y = bp[(4 * s + 1) * HID];   // B[k=4s+2*half+1][n]
                c = __builtin_amdgcn_wmma_f32_16x16x4_f32(
                        /*neg_a=*/false, a[s], /*neg_b=*/false, b,
                        /*c_mod=*/(short)0, c, /*reuse_a=*/false, /*reuse_b=*/false);
            }
            #pragma unroll
            for (int v = 0; v < 8; ++v)
                t[(size_t)(base + v + 8 * half) * HID + ncol + ml] = c[v];
        }
    } else {
        // scalar tail (never taken for N=50000, kept for generality; wave-uniform branch)
        for (int r = base; r < N; ++r)
            for (int j = lane; j < HID; j += 32) {
                float acc = 0.f;
                for (int k = 0; k < HID; ++k)
                    acc += h1[(size_t)r * HID + k] * W2[(size_t)k * HID + j];
                t[(size_t)r * HID + j] = acc;
            }
    }
}

// ---------------------------------------------------------------- acc[n][:] = dinv[n]^2 * t[n][:]   (self loop, no atomics)
__global__ void k_l2_init(const float* __restrict__ t, const float* __restrict__ dinv,
                          float* __restrict__ acc, int N) {
    int gid = blockIdx.x * blockDim.x + threadIdx.x;
    if (gid >= N * HID) return;
    int n = gid >> 6;
    float dn = dinv[n];
    acc[gid] = dn * dn * t[gid];
}

// ---------------------------------------------------------------- layer-2 wide edge scatter: 16 threads/edge, float4 each
__global__ void k_l2_scatter(const int* __restrict__ row, const int* __restrict__ col,
                             const float* __restrict__ t, const float* __restrict__ dinv,
                             float* __restrict__ acc, int E) {
    int gid = blockIdx.x * blockDim.x + threadIdx.x;
    int e = gid >> 4;
    if (e >= E) return;
    int q = gid & 15;
    int r = row[e], c = col[e];
    float nrm = dinv[r] * dinv[c];
    const float4* tr = (const float4*)(t + (size_t)r * HID);
    float4 v = tr[q];
    float* a = acc + (size_t)c * HID + q * 4;
    atomicAdd(a + 0, nrm * v.x);
    atomicAdd(a + 1, nrm * v.y);
    atomicAdd(a + 2, nrm * v.z);
    atomicAdd(a + 3, nrm * v.w);
}

// ---------------------------------------------------------------- fused relu(acc + b2) + global mean-pool partials
__global__ void k_pool(const float* __restrict__ acc, const float* __restrict__ b2,
                       float* __restrict__ gacc, int N) {
    __shared__ float sh[HID];
    int tid = threadIdx.x;
    int j = tid & 63;
    int rg = tid >> 6;              // 0..3 (256 threads)
    float bj = b2[j];
    float local = 0.f;
    for (int n = blockIdx.x * 4 + rg; n < N; n += gridDim.x * 4)
        local += fmaxf(acc[(size_t)n * HID + j] + bj, 0.f);
    if (tid < HID) sh[tid] = 0.f;
    __syncthreads();
    atomicAdd(&sh[j], local);
    __syncthreads();
    if (tid < HID) atomicAdd(&gacc[tid], sh[tid]);
}

// ---------------------------------------------------------------- head: g = mean, logits = g@Wfc + bfc, log_softmax
__global__ void k_head(const float* __restrict__ gacc, const float* __restrict__ Wfc,
                       const float* __restrict__ bfc, float* __restrict__ out, float invN) {
    __shared__ float g[HID];
    __shared__ float logits[4];
    int tid = threadIdx.x;
    if (tid < HID) g[tid] = gacc[tid] * invN;
    __syncthreads();
    if (tid < 4) {
        float a = bfc[tid];
        for (int j = 0; j < HID; ++j) a += g[j] * Wfc[j * 4 + tid];
        logits[tid] = a;
    }
    __syncthreads();
    if (tid == 0) {
        float m = logits[0];
        for (int v = 1; v < 4; ++v) m = fmaxf(m, logits[v]);
        float sum = 0.f;
        for (int v = 0; v < 4; ++v) sum += expf(logits[v] - m);
        float lse = logf(sum);
        for (int v = 0; v < 4; ++v) out[v] = logits[v] - m - lse;
    }
}

extern "C" void kernel_launch(void* const* d_in, const int* in_sizes, int n_in,
                              void* d_out, int out_size, void* d_ws, size_t ws_size,
                              hipStream_t stream) {
    const float* x   = (const float*)d_in[0];
    const int*   ei  = (const int*)d_in[1];    // [2, E], row-major
    const float* W1  = (const float*)d_in[2];
    const float* b1  = (const float*)d_in[3];
    const float* W2  = (const float*)d_in[4];
    const float* b2  = (const float*)d_in[5];
    const float* Wfc = (const float*)d_in[6];
    const float* bfc = (const float*)d_in[7];
    float* out = (float*)d_out;

    const int N = in_sizes[0];          // x is [N,1]
    const int E = in_sizes[1] / 2;
    const int* row = ei;
    const int* col = ei + E;

    // workspace layout (floats), base offsets padded to 16-float alignment
    size_t Np = ((size_t)N + 15) & ~(size_t)15;
    float* ws   = (float*)d_ws;
    float* dinv = ws;                   // N   (deg, then deg^{-1/2})
    float* s    = ws + Np;              // N
    float* h1   = ws + 2 * Np;          // 64N (reused as `acc` after GEMM)
    float* t    = h1 + (size_t)HID * Np;// 64N
    float* gacc = t + (size_t)HID * Np; // 64
    float* acc  = h1;

    const int T = 256;
    k_init<<<(N + T - 1) / T, T, 0, stream>>>(dinv, s, gacc, N);
    k_degree<<<(E + T - 1) / T, T, 0, stream>>>(col, dinv, E);
    k_dinv<<<(N + T - 1) / T, T, 0, stream>>>(dinv, N);
    k_l1_scatter<<<(E + T - 1) / T, T, 0, stream>>>(row, col, x, dinv, s, E);

    int NH = N * HID;
    k_l1_finish<<<(NH + T - 1) / T, T, 0, stream>>>(s, dinv, x, W1, b1, h1, N);

    int ntiles = (N + 15) / 16;                 // one wave per 16-row tile
    int gblocks = (ntiles * 32 + T - 1) / T;
    k_gemm_wmma<<<gblocks, T, 0, stream>>>(h1, W2, t, N);

    k_l2_init<<<(NH + T - 1) / T, T, 0, stream>>>(t, dinv, acc, N);

    long long ET = (long long)E * 16;
    k_l2_scatter<<<(int)((ET + T - 1) / T), T, 0, stream>>>(row, col, t, dinv, acc, E);

    k_pool<<<512, 256, 0, stream>>>(acc, b2, gacc, N);
    k_head<<<1, 64, 0, stream>>>(gacc, Wfc, bfc, out, 1.0f / (float)N);
}